// SparseInverseConv3d_2594160247595
// MI455X (gfx1250) — compile-verified
//
#include <hip/hip_runtime.h>

typedef __attribute__((ext_vector_type(2))) float v2f;
typedef __attribute__((ext_vector_type(4))) float v4f;
typedef __attribute__((ext_vector_type(8))) float v8f;

#define N_IN   100000
#define N_OUT  800000
#define KOFF   27
#define C_IN   32
#define C_OUT  64
#define TILES_P (N_IN / 16)          // 6250, exact
#define WAVES_PER_BLOCK 8
#define THREADS (WAVES_PER_BLOCK * 32)

// ---------------------------------------------------------------------------
// out[i, c] = bias[c] for all 800k rows (replaces zeros + bias add).
// 4 floats per thread, fully coalesced b128 stores.
// ---------------------------------------------------------------------------
__global__ __launch_bounds__(256)
void bias_init_kernel(const float* __restrict__ bias, float* __restrict__ out) {
    const long long total = (long long)N_OUT * C_OUT;
    const long long base = ((long long)blockIdx.x * blockDim.x + threadIdx.x) * 4;
    if (base >= total) return;
    const int c = (int)(base & (C_OUT - 1));   // aligned to 4, C_OUT pow2
    v4f b;
    b.x = bias[c + 0]; b.y = bias[c + 1]; b.z = bias[c + 2]; b.w = bias[c + 3];
    *(v4f*)(out + base) = b;
}

// ---------------------------------------------------------------------------
// One wave = one (k, 16-row P-tile). Computes a 16x64 contribution tile with
// 8 K-steps x 4 N-tiles of v_wmma_f32_16x16x4_f32, then atomic-scatter-adds
// it into the output via the rulebook's out_idx.
// ---------------------------------------------------------------------------
__global__ __launch_bounds__(THREADS)
void scatter_gemm_kernel(const float* __restrict__ feats,
                         const float* __restrict__ weight,
                         const int*   __restrict__ bwd,
                         const int*   __restrict__ in_idx,
                         const int*   __restrict__ out_idx,
                         float*       __restrict__ out) {
    const int wave = blockIdx.x * WAVES_PER_BLOCK + (threadIdx.x >> 5);
    const int totalTiles = KOFF * TILES_P;
    if (wave >= totalTiles) return;            // uniform per wave: EXEC stays all-1s

    const int lane = threadIdx.x & 31;
    const int m    = lane & 15;                // A: row M / B,C,D: column N (within tile)
    const int kh   = lane >> 4;                // K-half select per 16x16x4 f32 layout

    const int k  = wave / TILES_P;
    const int p0 = (wave - k * TILES_P) * 16;

    // A-matrix source row after the bwd permutation composed with the rulebook gather:
    //   gathered[k, p] = feats[ bwd[ in_idx[k, p] ] ]
    const int arow = bwd[in_idx[k * N_IN + p0 + m]];

    // A lane layout (16x4 f32): v0 = K = 2*kh + 0, v1 = K = 2*kh + 1
    const float* aptr  = feats + (long long)arow * C_IN + kh * 2;
    // B lane layout (4x16 f32): v0 = row K = 2*kh + 0, col m; v1 = row K = 2*kh + 1
    const float* wbase = weight + ((long long)k * C_IN + kh * 2) * C_OUT + m;

    v8f acc0 = {}, acc1 = {}, acc2 = {}, acc3 = {};

#pragma unroll
    for (int kk = 0; kk < C_IN; kk += 4) {
        v2f a = *(const v2f*)(aptr + kk);      // contiguous 8B gather from feature row
        const float* wp = wbase + kk * C_OUT;
        v2f b0, b1, b2, b3;
        b0.x = wp[0];  b0.y = wp[C_OUT + 0];
        b1.x = wp[16]; b1.y = wp[C_OUT + 16];
        b2.x = wp[32]; b2.y = wp[C_OUT + 32];
        b3.x = wp[48]; b3.y = wp[C_OUT + 48];
        acc0 = __builtin_amdgcn_wmma_f32_16x16x4_f32(false, a, false, b0, (short)0, acc0, false, false);
        acc1 = __builtin_amdgcn_wmma_f32_16x16x4_f32(false, a, false, b1, (short)0, acc1, false, false);
        acc2 = __builtin_amdgcn_wmma_f32_16x16x4_f32(false, a, false, b2, (short)0, acc2, false, false);
        acc3 = __builtin_amdgcn_wmma_f32_16x16x4_f32(false, a, false, b3, (short)0, acc3, false, false);
    }

    // C/D layout: VGPR r holds row M = 8*kh + r, column N = m (+ 16*j per N-tile).
    const int* oix = out_idx + k * N_IN + p0;
#pragma unroll
    for (int r = 0; r < 8; ++r) {
        const int orow = oix[kh * 8 + r];
        float* op = out + (long long)orow * C_OUT + m;
        atomicAdd(op + 0,  acc0[r]);
        atomicAdd(op + 16, acc1[r]);
        atomicAdd(op + 32, acc2[r]);
        atomicAdd(op + 48, acc3[r]);
    }
}

extern "C" void kernel_launch(void* const* d_in, const int* in_sizes, int n_in,
                              void* d_out, int out_size, void* d_ws, size_t ws_size,
                              hipStream_t stream) {
    const float* feats   = (const float*)d_in[0];   // [N_IN, C_IN]
    const float* weight  = (const float*)d_in[1];   // [K, C_IN, C_OUT]
    const float* bias    = (const float*)d_in[2];   // [C_OUT]
    const int*   bwd     = (const int*)d_in[3];     // [N_IN]
    const int*   in_idx  = (const int*)d_in[4];     // [K, N_IN]
    const int*   out_idx = (const int*)d_in[5];     // [K, N_IN]
    // d_in[6] = n_out scalar (compile-time constant here)
    float* out = (float*)d_out;                     // [N_OUT, C_OUT]

    {
        const long long totalV4 = (long long)N_OUT * C_OUT / 4;
        const int blocks = (int)((totalV4 + 255) / 256);
        bias_init_kernel<<<blocks, 256, 0, stream>>>(bias, out);
    }
    {
        const int totalTiles = KOFF * TILES_P;
        const int blocks = (totalTiles + WAVES_PER_BLOCK - 1) / WAVES_PER_BLOCK;
        scatter_gemm_kernel<<<blocks, THREADS, 0, stream>>>(feats, weight, bwd,
                                                            in_idx, out_idx, out);
    }
}